// Channel_Att_Bridge_32212254720180
// MI455X (gfx1250) — compile-verified
//
#include <hip/hip_runtime.h>
#include <math.h>

#define CSUM 2016   // sum(C_LIST) = 32+64+128+256+512+512+512
#define BATCH 32

typedef float v2f __attribute__((ext_vector_type(2)));
typedef float v4f __attribute__((ext_vector_type(4)));
typedef float v8f __attribute__((ext_vector_type(8)));

struct Ptrs7 { const float* p[7]; };

// ---------------------------------------------------------------------------
// Stage 0: zero the pooled-sum accumulator
// ---------------------------------------------------------------------------
__global__ void zero_kernel(float* p, int n) {
    int i = blockIdx.x * blockDim.x + threadIdx.x;
    if (i < n) p[i] = 0.0f;
}

// ---------------------------------------------------------------------------
// Stage 1: pooling. grid = (7*32, 8 segments), 512 threads.
// Single-touch 525 MB stream -> non-temporal b128 loads so we don't evict the
// small reused tables (g, weights) from L2.
// Thread t always owns channel group 4*(t % (c/4)) (all c are powers of 2).
// ---------------------------------------------------------------------------
__global__ __launch_bounds__(512)
void pool_kernel(Ptrs7 ts, float* __restrict__ pooled) {
    constexpr int   C[7]    = {32, 64, 128, 256, 512, 512, 512};
    constexpr int   LC4[7]  = {1 << 19, 1 << 18, 1 << 17, 1 << 16, 1 << 15, 1 << 13, 1 << 11};
    constexpr int   OFF[7]  = {0, 32, 96, 224, 480, 992, 1504};
    constexpr float INVL[7] = {1.f/65536.f, 1.f/16384.f, 1.f/4096.f, 1.f/1024.f,
                               1.f/256.f, 1.f/64.f, 1.f/16.f};

    const int lvl = blockIdx.x >> 5;   // / 32
    const int b   = blockIdx.x & 31;
    const int seg = blockIdx.y;

    const int lc4  = LC4[lvl];          // float4s per (b) slab
    const int per  = lc4 >> 3;          // float4s per segment (>=256, mult of c4)
    const int base = seg * per;
    const int c4   = C[lvl] >> 2;       // channel groups of 4

    const v4f* __restrict__ src =
        (const v4f*)ts.p[lvl] + (size_t)b * (size_t)lc4;

    const int t = threadIdx.x;
    v4f acc = {0.f, 0.f, 0.f, 0.f};
    for (int j = base + t; j < base + per; j += 512) {
        v4f v = __builtin_nontemporal_load(src + j);   // NT streaming b128 read
        acc += v;
    }

    __shared__ v4f sdata[512];
    sdata[t] = acc;
    __syncthreads();

    if (t < c4) {
        v4f s = {0.f, 0.f, 0.f, 0.f};
        for (int k = t; k < 512; k += c4) {
            s += sdata[k];
        }
        const float inv = INVL[lvl];
        float* dst = pooled + (size_t)b * CSUM + OFF[lvl] + 4 * t;
        atomicAdd(dst + 0, s[0] * inv);
        atomicAdd(dst + 1, s[1] * inv);
        atomicAdd(dst + 2, s[2] * inv);
        atomicAdd(dst + 3, s[3] * inv);
    }
}

// ---------------------------------------------------------------------------
// Stage 2: Conv1d(1,1,k=3,pad=1) along the channel axis, per batch row
// ---------------------------------------------------------------------------
__global__ void conv_kernel(const float* __restrict__ pooled,
                            const float* __restrict__ cw,
                            const float* __restrict__ cb,
                            float* __restrict__ att) {
    int i = blockIdx.x * blockDim.x + threadIdx.x;
    if (i >= BATCH * CSUM) return;
    int b = i / CSUM;
    int c = i - b * CSUM;
    const float* row = pooled + (size_t)b * CSUM;
    float xl = (c > 0)        ? row[c - 1] : 0.0f;
    float xc = row[c];
    float xr = (c < CSUM - 1) ? row[c + 1] : 0.0f;
    att[i] = cw[0] * xl + cw[1] * xc + cw[2] * xr + cb[0];
}

// ---------------------------------------------------------------------------
// Stage 3: dense GEMM (32 x 2016 x 2016) via V_WMMA_F32_16X16X4_F32
// + bias + IntensityGate (silu). One wave per 16x16 output tile.
// grid = 63 blocks * 4 waves = 252 waves = 2 rowTiles * 126 colTiles.
// A VGPR layout (16x4 f32): lane = (K/2)*16 + M, vgpr v holds K%2==v.
// B (4x16): lane = (K/2)*16 + N symmetric. D: row = v + 8*(lane/16), col = lane%16.
// ---------------------------------------------------------------------------
__global__ __launch_bounds__(128)
void gemm_silu_kernel(const float* __restrict__ att,
                      Ptrs7 Ws, Ptrs7 Bs,
                      const float* __restrict__ thr,
                      const float* __restrict__ enh,
                      float* __restrict__ g) {
    constexpr int TILE_CUM[8] = {0, 2, 6, 14, 30, 62, 94, 126};
    constexpr int OFF[7]      = {0, 32, 96, 224, 480, 992, 1504};

    const int wave    = (blockIdx.x * 128 + (int)threadIdx.x) >> 5; // 0..251
    const int rowTile = wave / 126;                                 // 0..1
    const int colTile = wave - rowTile * 126;                       // 0..125

    int lvl = 0;
#pragma unroll
    for (int i = 1; i < 7; ++i)
        if (colTile >= TILE_CUM[i]) lvl = i;
    const int tIn = colTile - TILE_CUM[lvl];

    const int lane  = threadIdx.x & 31;
    const int mlo   = lane & 15;   // M for A, N for B/D
    const int khalf = lane >> 4;   // K pair select

    const float* __restrict__ arow = att       + (size_t)(rowTile * 16 + mlo) * CSUM + 2 * khalf;
    const float* __restrict__ brow = Ws.p[lvl] + (size_t)(tIn     * 16 + mlo) * CSUM + 2 * khalf;

    v8f acc = {0.f, 0.f, 0.f, 0.f, 0.f, 0.f, 0.f, 0.f};
    for (int kk = 0; kk < CSUM; kk += 4) {
        v2f a    = *(const v2f*)(arow + kk);
        v2f bmat = *(const v2f*)(brow + kk);
        // D = A(16x4,f32) x B(4x16,f32) + C, full f32 precision
        acc = __builtin_amdgcn_wmma_f32_16x16x4_f32(
            /*neg_a=*/false, a, /*neg_b=*/false, bmat,
            /*c_mod=*/(short)0, acc, /*reuse_a=*/false, /*reuse_b=*/false);
    }

    const float bias = Bs.p[lvl][tIn * 16 + mlo];
    const float T = thr[0];
    const float E = enh[0];
    const int ncol = OFF[lvl] + tIn * 16 + mlo;

#pragma unroll
    for (int v = 0; v < 8; ++v) {
        int m = rowTile * 16 + v + 8 * khalf;     // batch row 0..31
        float x = E * (acc[v] + bias - T);
        float s = 1.0f / (1.0f + __expf(-x));
        g[(size_t)m * CSUM + ncol] = x * s;
    }
}

// ---------------------------------------------------------------------------
// Stage 4: broadcast g(32,2016) to all 7 output tensors.
// 525 MB single-touch store stream -> non-temporal b128 stores; g loads stay
// regular-temporal (258 KB, heavily reused, cache-resident).
// ---------------------------------------------------------------------------
__global__ __launch_bounds__(256)
void bcast_kernel(const float* __restrict__ g, v4f* __restrict__ out) {
    // per-level output size in float4: B*l*c/4
    constexpr long CUM[8] = {0L, 16777216L, 25165824L, 29360128L,
                             31457280L, 32505856L, 32768000L, 32833536L};
    constexpr int SH[7]  = {19, 18, 17, 16, 15, 13, 11};      // log2(l*c/4)
    constexpr int C4M[7] = {7, 15, 31, 63, 127, 127, 127};    // c/4 - 1
    constexpr int OFF[7] = {0, 32, 96, 224, 480, 992, 1504};

    const long total  = CUM[7];
    const long stride = (long)gridDim.x * blockDim.x;
    for (long i = (long)blockIdx.x * blockDim.x + threadIdx.x; i < total; i += stride) {
        int lvl;
        if (i >= CUM[4])      lvl = (i >= CUM[6]) ? 6 : ((i >= CUM[5]) ? 5 : 4);
        else if (i >= CUM[2]) lvl = (i >= CUM[3]) ? 3 : 2;
        else                  lvl = (i >= CUM[1]) ? 1 : 0;

        long r  = i - CUM[lvl];
        int  b  = (int)(r >> SH[lvl]);
        int  c4 = ((int)r) & C4M[lvl];
        v4f v = *((const v4f*)(g + (size_t)b * CSUM + OFF[lvl]) + c4);
        __builtin_nontemporal_store(v, out + i);   // NT streaming b128 write
    }
}

// ---------------------------------------------------------------------------
extern "C" void kernel_launch(void* const* d_in, const int* in_sizes, int n_in,
                              void* d_out, int out_size, void* d_ws, size_t ws_size,
                              hipStream_t stream) {
    static const int C[7] = {32, 64, 128, 256, 512, 512, 512};

    Ptrs7 Ts, WW, BB;
    for (int i = 0; i < 7; ++i) Ts.p[i] = (const float*)d_in[i];
    const float* cw = (const float*)d_in[7];
    const float* cb = (const float*)d_in[8];

    const float *thr, *enh;
    if (n_in >= 25) {
        // lists flattened into separate inputs
        for (int i = 0; i < 7; ++i) WW.p[i] = (const float*)d_in[9 + i];
        for (int i = 0; i < 7; ++i) BB.p[i] = (const float*)d_in[16 + i];
        thr = (const float*)d_in[23];
        enh = (const float*)d_in[24];
    } else {
        // lists passed as concatenated buffers
        const float* Wall = (const float*)d_in[9];
        const float* ball = (const float*)d_in[10];
        size_t wo = 0, bo = 0;
        for (int i = 0; i < 7; ++i) {
            WW.p[i] = Wall + wo; wo += (size_t)C[i] * CSUM;
            BB.p[i] = ball + bo; bo += (size_t)C[i];
        }
        thr = (const float*)d_in[11];
        enh = (const float*)d_in[12];
    }

    float* pooled = (float*)d_ws;               // 32*2016 floats
    float* attb   = pooled + BATCH * CSUM;      // 32*2016 floats
    float* gb     = attb   + BATCH * CSUM;      // 32*2016 floats

    const int nPool = BATCH * CSUM;
    zero_kernel<<<(nPool + 255) / 256, 256, 0, stream>>>(pooled, nPool);
    pool_kernel<<<dim3(7 * BATCH, 8), 512, 0, stream>>>(Ts, pooled);
    conv_kernel<<<(nPool + 255) / 256, 256, 0, stream>>>(pooled, cw, cb, attb);
    gemm_silu_kernel<<<63, 128, 0, stream>>>(attb, WW, BB, thr, enh, gb);
    bcast_kernel<<<16384, 256, 0, stream>>>(gb, (v4f*)d_out);

    (void)in_sizes; (void)out_size; (void)ws_size;
}